// DensityFunction_84482006712774
// MI455X (gfx1250) — compile-verified
//
#include <hip/hip_runtime.h>
#include <hip/hip_bf16.h>

// CDNA5 / gfx1250 implementation of the WFN density + FD-gradient kernel.
// Core math: density[n] = x_n^T S x_n with S = symmetrized dm (256x256 fp32),
// evaluated 7x per point (center + 6 finite-difference shifts).
// The quadratic form is executed as V_WMMA_F32_16X16X4_F32 GEMM tiles:
//   Y[256,16] = S[256,256] @ X[256,16]  per 16-point tile, per variant,
// then density[n] = sum_p X[p,n] * Y[p,n].

typedef __attribute__((ext_vector_type(2))) float v2f;
typedef __attribute__((ext_vector_type(8))) float v8f;

#define PP 256   // number of primitives (fixed by the problem)
#define NT 16    // points per wave tile (WMMA N dimension)

// Packed monomial powers for the 10-row Cartesian angular-momentum table:
// bits [1:0]=pow_x, [3:2]=pow_y, [5:4]=pow_z
__constant__ int kPwPacked[10] = {
    0,          // (0,0,0)
    1,          // (1,0,0)
    4,          // (0,1,0)
    16,         // (0,0,1)
    2,          // (2,0,0)
    8,          // (0,2,0)
    32,         // (0,0,2)
    5,          // (1,1,0)
    17,         // (1,0,1)
    20          // (0,1,1)
};

__device__ __forceinline__ float powcomp(float v, int e) {
    // e in {0,1,2}: 1, v, v*v  (matches reference's select, avoids pow())
    float r = (e == 0) ? 1.0f : v;
    return (e == 2) ? v * v : r;
}

// ---------------------------------------------------------------------------
// Kernel 1: S[i][j] = dm[max(i,j)][min(i,j)]  (tril + tril^T + diag)
// ---------------------------------------------------------------------------
__global__ void symmetrize_kernel(const float* __restrict__ dm,
                                  float* __restrict__ S) {
    int i = blockIdx.x;       // row    (256 blocks)
    int j = threadIdx.x;      // column (256 threads)
    int hi = i > j ? i : j;
    int lo = i > j ? j : i;
    S[i * PP + j] = dm[hi * PP + lo];
}

// ---------------------------------------------------------------------------
// Kernel 2: one wave32 per 16-point tile; 7 density evaluations per point.
// ---------------------------------------------------------------------------
__global__ __launch_bounds__(32)
void density_grad_kernel(const float* __restrict__ coords,       // [N,3]
                         const float* __restrict__ atom_coords,  // [A,3]
                         const int*   __restrict__ centers,      // [P]
                         const float* __restrict__ alpha,        // [P]
                         const int*   __restrict__ sym,          // [P]
                         const float* __restrict__ S,            // [P,P] symmetrized
                         float* __restrict__ outP,               // [N]
                         float* __restrict__ outG,               // [N]
                         int N) {
    __shared__ float sAx[PP], sAy[PP], sAz[PP], sAl[PP];
    __shared__ int   sPw[PP];
    __shared__ float sX[PP * NT];   // X tile, row-major [p][n], 16 KB

    const int lane = threadIdx.x;   // 0..31 (wave32)
    const int nl   = lane & 15;     // point column within tile
    const int hh   = lane >> 4;     // lane half (WMMA K/M split)
    const int n0   = blockIdx.x * NT;

    // ---- stage per-primitive constants into LDS (variant-invariant) ----
    #pragma unroll
    for (int i = 0; i < PP / 32; ++i) {
        int p = i * 32 + lane;
        int c = centers[p];
        sAx[p] = atom_coords[c * 3 + 0];
        sAy[p] = atom_coords[c * 3 + 1];
        sAz[p] = atom_coords[c * 3 + 2];
        sAl[p] = alpha[p];
        sPw[p] = kPwPacked[sym[p]];
    }

    // point coordinates for this lane's column (clamped for ragged tails)
    int nIdx = n0 + nl;
    if (nIdx >= N) nIdx = N - 1;
    const float cx = coords[nIdx * 3 + 0];
    const float cy = coords[nIdx * 3 + 1];
    const float cz = coords[nIdx * 3 + 2];

    __syncthreads();

    const float dh = 1e-8f;
    float dens[7];

    for (int v = 0; v < 7; ++v) {
        // finite-difference shifted coordinates (v=0 is the center eval)
        float px = cx + ((v == 1) ? dh : (v == 2) ? -dh : 0.0f);
        float py = cy + ((v == 3) ? dh : (v == 4) ? -dh : 0.0f);
        float pz = cz + ((v == 5) ? dh : (v == 6) ? -dh : 0.0f);

        // ---- build basis-value tile X[256][16] in LDS ----
        // lane handles column nl, primitives p = 2*i + hh (conflict-free banks)
        #pragma unroll 4
        for (int i = 0; i < PP / 2; ++i) {
            int p = 2 * i + hh;
            float vx = px - sAx[p];
            float vy = py - sAy[p];
            float vz = pz - sAz[p];
            float d2 = vx * vx + vy * vy + vz * vz;
            int pw = sPw[p];
            float sh = powcomp(vx, pw & 3) *
                       powcomp(vy, (pw >> 2) & 3) *
                       powcomp(vz, (pw >> 4) & 3);
            sX[p * NT + nl] = sh * __expf(-sAl[p] * d2);
        }
        __syncthreads();

        // ---- Y = S @ X via V_WMMA_F32_16X16X4_F32, fused dot with X ----
        // A tile (16x4 fp32): lane m = nl, K-pair selected by lane half hh.
        const float2* __restrict__ A2 =
            (const float2*)(S + (size_t)nl * PP + 2 * hh);   // 8B aligned
        // B tile (4x16 fp32): VGPR0 rows {k+hh}, VGPR1 rows {k+2+hh}, col nl.
        const float* __restrict__ Xb = sX + hh * NT + nl;
        // Dot operand: C-layout row m = m0 + r + 8*hh, col nl.
        const float* __restrict__ Xd = sX + (8 * hh) * NT + nl;

        float dsum = 0.0f;
        #pragma unroll 1
        for (int mt = 0; mt < PP / 16; ++mt) {
            v8f acc = {};
            const float2* __restrict__ Arow = A2 + (size_t)(mt * 16) * (PP / 2);
            #pragma unroll 8
            for (int k = 0; k < PP; k += 4) {
                v2f a;
                float2 av = Arow[k >> 1];          // global_load_b64 (L2-hot S)
                a.x = av.x; a.y = av.y;
                v2f b;
                b.x = Xb[k * NT];                  // ds_load pair, no conflicts
                b.y = Xb[(k + 2) * NT];
                acc = __builtin_amdgcn_wmma_f32_16x16x4_f32(
                    /*neg_a=*/false, a, /*neg_b=*/false, b,
                    /*c_mod=*/(short)0, acc,
                    /*reuse_a=*/false, /*reuse_b=*/false);
            }
            const float* __restrict__ Xdm = Xd + (mt * 16) * NT;
            #pragma unroll
            for (int r = 0; r < 8; ++r)
                dsum += acc[r] * Xdm[r * NT];
        }
        // combine lane halves (rows m%16 in 0..7 vs 8..15)
        dsum += __shfl_down(dsum, 16, 32);
        dens[v] = dsum;            // valid in lanes 0..15
        __syncthreads();           // protect sX before next variant overwrites
    }

    if (lane < 16 && (n0 + nl) < N) {
        outP[n0 + nl] = dens[0];
        const float inv = 1.0f / (2.0f * dh);
        float gx = (dens[1] - dens[2]) * inv;
        float gy = (dens[3] - dens[4]) * inv;
        float gz = (dens[5] - dens[6]) * inv;
        outG[n0 + nl] = sqrtf(gx * gx + gy * gy + gz * gz);
    }
}

// ---------------------------------------------------------------------------
extern "C" void kernel_launch(void* const* d_in, const int* in_sizes, int n_in,
                              void* d_out, int out_size, void* d_ws, size_t ws_size,
                              hipStream_t stream) {
    const float* coords      = (const float*)d_in[0];   // [N,3]
    const float* atom_coords = (const float*)d_in[1];   // [A,3]
    const int*   centers     = (const int*)d_in[2];     // [P]
    const float* expo        = (const float*)d_in[3];   // [P]
    const int*   sym         = (const int*)d_in[4];     // [P]
    const float* dm          = (const float*)d_in[5];   // [P,P]

    const int N = in_sizes[0] / 3;
    float* S = (float*)d_ws;                // 256*256*4 = 256 KB scratch

    hipLaunchKernelGGL(symmetrize_kernel, dim3(PP), dim3(PP), 0, stream, dm, S);

    float* outP = (float*)d_out;
    float* outG = (float*)d_out + N;
    const int ntiles = (N + NT - 1) / NT;
    hipLaunchKernelGGL(density_grad_kernel, dim3(ntiles), dim3(32), 0, stream,
                       coords, atom_coords, centers, expo, sym, S,
                       outP, outG, N);
}